// TAMCaD_70471823392788
// MI455X (gfx1250) — compile-verified
//
#include <hip/hip_runtime.h>

// ---------------------------------------------------------------------------
// TAMCaD group-vs-group attention for MI455X (gfx1250, wave32, WMMA).
//
// Shapes (fixed by reference): B=4, Gq=16, D=128, T=4096, H=8, dk=16, Gkv=32.
// Memory-bound (AI ~5 FLOP/B, ~840 MB total -> ~36us floor @ 23.3 TB/s).
// Strategy: one pass over HBM; GLOBAL_LOAD_ASYNC_TO_LDS staging (ASYNCcnt)
// for the t-major -> matrix-striped layout transform; WMMA f32 16x16x4 for
// the per-timestep 16x16x32 matmuls; DPP16 row_ror butterflies (no LDS) for
// the softmax row reductions.
// ---------------------------------------------------------------------------

typedef __attribute__((ext_vector_type(2))) float v2f;  // A/B frag: 64 f32 / 32 lanes
typedef __attribute__((ext_vector_type(8))) float v8f;  // C/D frag: 256 f32 / 32 lanes

#define B_   4
#define G_   16      // query groups
#define GK_  32      // kv groups (reg + instant)
#define D_   128
#define T_   4096
#define H_   8
#define DK_  16
#define TT_  32      // timesteps per workgroup
#define NT_  512     // 16 waves of 32

#define QROW 17              // dk row padded (odd -> conflict-free col reads)
#define QT   (G_ * QROW)     // 272 floats per t for Q (also x scratch)
#define KT   (GK_ * QROW)    // 544 floats per t for K / V
#define AROW 33              // attn row padded (16+16 cols + 1)
#define APW  (G_ * AROW)     // 528 floats per-wave attn scratch

// dynamic LDS partition (floats)
#define OFF_Q 0
#define OFF_K (TT_ * QT)                   // 8704
#define OFF_V (OFF_K + TT_ * KT)           // 26112
#define OFF_A (OFF_V + TT_ * KT)           // 43520
#define SM_FLOATS (OFF_A + 16 * APW)       // 51968 floats = 207,872 B dynamic LDS

typedef __attribute__((address_space(1))) int  as1_int;  // global
typedef __attribute__((address_space(3))) int  as3_int;  // LDS

// ---- DPP16 row-rotate butterflies (reduce across a 16-lane row, VALU-only) ----
template <int CTRL>
__device__ __forceinline__ float dpp_rot(float x) {
    int b = __builtin_bit_cast(int, x);
    int r = __builtin_amdgcn_update_dpp(b, b, CTRL, 0xf, 0xf, false);
    return __builtin_bit_cast(float, r);
}
__device__ __forceinline__ float row_max16(float x) {
    x = fmaxf(x, dpp_rot<0x128>(x));   // row_ror:8
    x = fmaxf(x, dpp_rot<0x124>(x));   // row_ror:4
    x = fmaxf(x, dpp_rot<0x122>(x));   // row_ror:2
    x = fmaxf(x, dpp_rot<0x121>(x));   // row_ror:1
    return x;
}
__device__ __forceinline__ float row_sum16(float x) {
    x += dpp_rot<0x128>(x);
    x += dpp_rot<0x124>(x);
    x += dpp_rot<0x122>(x);
    x += dpp_rot<0x121>(x);
    return x;
}

__global__ void tamcad_zero_f4(float4* __restrict__ p, int n4) {
    int i = blockIdx.x * blockDim.x + threadIdx.x;
    if (i < n4) p[i] = make_float4(0.f, 0.f, 0.f, 0.f);
}

__global__ __launch_bounds__(NT_) void tamcad_attn_wmma(
    const float* __restrict__ q, const float* __restrict__ k,
    const float* __restrict__ v, const float* __restrict__ ki,
    const float* __restrict__ vi, float* __restrict__ outx,
    float* __restrict__ outa)
{
    extern __shared__ float sm[];
    float* smQ = sm + OFF_Q;   // [t][g*17+c]; reused as x scratch
    float* smK = sm + OFF_K;   // [t][gk*17+c], gk 0..15 = k, 16..31 = k_instant
    float* smV = sm + OFF_V;   // [t][gk*17+c]
    float* smA = sm + OFF_A;   // per-wave [q*33 + gk] transpose scratch

    const int tid  = threadIdx.x;
    const int lane = tid & 31;
    const int wid  = tid >> 5;
    const int bid  = blockIdx.x;
    const int t0   = (bid & 127) * TT_;  // T_/TT_ = 128 tiles
    const int h    = (bid >> 7) & 7;
    const int b    = bid >> 10;

    // ---- Stage: 1280 rows x 32 t into LDS ----
    // rows: [0,256)=Q [256,512)=K [512,768)=K_inst [768,1024)=V [1024,1280)=V_inst
#if __has_builtin(__builtin_amdgcn_global_load_async_to_lds_b32)
    // Async direct-to-LDS: row is wave-uniform (SALU decode), lane supplies t
    // -> 128B coalesced global segments, per-lane LDS scatter, ASYNCcnt-tracked.
    for (int j = 0; j < 80; ++j) {
        int row = wid * 80 + j;           // 0..1279
        int rr  = row & 255;
        int g = rr >> 4, c = rr & 15;
        const float* src; float* dstp; int grp = g; int stride;
        if      (row <  256) { src = q;  dstp = smQ; stride = QT; }
        else if (row <  512) { src = k;  dstp = smK; stride = KT; }
        else if (row <  768) { src = ki; dstp = smK; stride = KT; grp = g + 16; }
        else if (row < 1024) { src = v;  dstp = smV; stride = KT; }
        else                 { src = vi; dstp = smV; stride = KT; grp = g + 16; }
        long gidx = (((long)(b * G_ + g) * D_) + h * DK_ + c) * T_ + t0 + lane;
        const float* gp = src + gidx;
        float* lp = dstp + lane * stride + grp * QROW + c;
        __builtin_amdgcn_global_load_async_to_lds_b32(
            (as1_int*)gp, (as3_int*)lp, 0, 0);
    }
#if __has_builtin(__builtin_amdgcn_s_wait_asynccnt)
    __builtin_amdgcn_s_wait_asynccnt(0);
#else
    asm volatile("s_wait_asynccnt 0" ::: "memory");
#endif
#else
    // Fallback: b128 load + LDS scatter via VGPRs.
    for (int i = 0; i < 20; ++i) {
        int ch  = tid + i * NT_;          // 0..10239
        int row = ch >> 3;
        int tc  = ch & 7;
        int rr  = row & 255;
        int g   = rr >> 4, c = rr & 15;
        const float* src; float* dstp; int grp = g;
        if      (row <  256) { src = q;  dstp = smQ; }
        else if (row <  512) { src = k;  dstp = smK; }
        else if (row <  768) { src = ki; dstp = smK; grp = g + 16; }
        else if (row < 1024) { src = v;  dstp = smV; }
        else                 { src = vi; dstp = smV; grp = g + 16; }
        long gidx = (((long)(b * G_ + g) * D_) + h * DK_ + c) * T_ + t0 + tc * 4;
        float4 val = *(const float4*)(src + gidx);
        int stride = (row < 256) ? QT : KT;
        int off    = grp * QROW + c;
        int tb     = tc * 4;
        dstp[(tb + 0) * stride + off] = val.x;
        dstp[(tb + 1) * stride + off] = val.y;
        dstp[(tb + 2) * stride + off] = val.z;
        dstp[(tb + 3) * stride + off] = val.w;
    }
#endif
    __syncthreads();

    // WMMA f32 16x16x4 fragment addressing (A 16x4: lanes 0-15 M, VGPRs K;
    // lane-half selects K pair; B 4x16 assumed mirror: lanes N, VGPR+half K).
    const int m16  = lane & 15;
    const int koff = (lane >> 4) * 2;
    const int half = lane >> 4;
    float* smAw = smA + wid * APW;

    for (int it = 0; it < 2; ++it) {
        const int t = wid * 2 + it;
        const float* sQt = smQ + t * QT;
        const float* sKt = smK + t * KT;
        const float* sVt = smV + t * KT;

        // ---- logits = (Q*1/sqrt(dk)) . K^T : two 16x16 accumulators ----
        v8f cr = {}; v8f ci = {};
#pragma unroll
        for (int kc = 0; kc < 4; ++kc) {
            const int kk = kc * 4 + koff;
            v2f a, br, bi;
            a.x  = sQt[m16 * QROW + kk] * 0.25f;   // fold scale into A frag
            a.y  = sQt[m16 * QROW + kk + 1] * 0.25f;
            br.x = sKt[m16 * QROW + kk];          br.y = sKt[m16 * QROW + kk + 1];
            bi.x = sKt[(m16 + 16) * QROW + kk];   bi.y = sKt[(m16 + 16) * QROW + kk + 1];
            cr = __builtin_amdgcn_wmma_f32_16x16x4_f32(false, a, false, br, (short)0, cr, false, false);
            ci = __builtin_amdgcn_wmma_f32_16x16x4_f32(false, a, false, bi, (short)0, ci, false, false);
        }

        // ---- softmax over 32 kv-groups: row (q) lives across a 16-lane row,
        //      reduced with DPP16 row_ror butterflies (no LDS traffic) ----
        v8f ar, ai;
#pragma unroll
        for (int r = 0; r < 8; ++r) {
            float mx = row_max16(fmaxf(cr[r], ci[r]));
            float er = __expf(cr[r] - mx);
            float ei = __expf(ci[r] - mx);
            float inv = __builtin_amdgcn_rcpf(row_sum16(er + ei));
            ar[r] = er * inv;
            ai[r] = ei * inv;
        }

        // ---- attn outputs (mean over h via f32 atomics) + C->A transpose ----
#pragma unroll
        for (int r = 0; r < 8; ++r) {
            const int qr = r + half * 8;
            smAw[qr * AROW + m16]      = ar[r];
            smAw[qr * AROW + 16 + m16] = ai[r];
            long o = (((long)(b * G_ + qr) * GK_) + m16) * T_ + t0 + t;
            atomicAdd(outa + o,            ar[r] * 0.125f);
            atomicAdd(outa + o + 16L * T_, ai[r] * 0.125f);
        }
        asm volatile("s_wait_dscnt 0" ::: "memory");  // wave-local LDS RAW fence

        // ---- x = attn . V : contraction over 32 kv-groups, 8 chained WMMAs ----
        v8f cx = {};
#pragma unroll
        for (int kc = 0; kc < 8; ++kc) {
            const int kk = kc * 4 + koff;
            v2f a, bv;
            a.x  = smAw[m16 * AROW + kk];     a.y  = smAw[m16 * AROW + kk + 1];
            bv.x = sVt[kk * QROW + m16];      bv.y = sVt[(kk + 1) * QROW + m16];
            cx = __builtin_amdgcn_wmma_f32_16x16x4_f32(false, a, false, bv, (short)0, cx, false, false);
        }

        // park x in this t's (now dead) Q region for a coalesced store phase
        float* sXt = smQ + t * QT;
#pragma unroll
        for (int r = 0; r < 8; ++r)
            sXt[(r + half * 8) * QROW + m16] = cx[r];
    }
    __syncthreads();

    // ---- cooperative x store: 256 rows x 32 t, float4 coalesced in t ----
    for (int i = 0; i < 4; ++i) {
        int ch  = tid + i * NT_;          // 0..2047
        int row = ch >> 3;                // (q, c)
        int tc  = ch & 7;
        int qr  = row >> 4, c = row & 15;
        float4 val;
        val.x = smQ[(tc * 4 + 0) * QT + qr * QROW + c];
        val.y = smQ[(tc * 4 + 1) * QT + qr * QROW + c];
        val.z = smQ[(tc * 4 + 2) * QT + qr * QROW + c];
        val.w = smQ[(tc * 4 + 3) * QT + qr * QROW + c];
        long o = (((long)(b * G_ + qr) * D_) + h * DK_ + c) * T_ + t0 + tc * 4;
        *(float4*)(outx + o) = val;
    }
}

extern "C" void kernel_launch(void* const* d_in, const int* in_sizes, int n_in,
                              void* d_out, int out_size, void* d_ws, size_t ws_size,
                              hipStream_t stream) {
    const float* q  = (const float*)d_in[0];
    const float* k  = (const float*)d_in[1];
    const float* v  = (const float*)d_in[2];
    const float* ki = (const float*)d_in[3];
    const float* vi = (const float*)d_in[4];
    float* outx = (float*)d_out;                          // (B, G*D, T)
    float* outa = outx + (long)B_ * G_ * D_ * T_;         // (B, G, GK, T)

    // zero the atomically-accumulated attention output
    int n4 = (B_ * G_ * GK_ * T_) / 4;                    // 2,097,152 float4s
    tamcad_zero_f4<<<(n4 + 255) / 256, 256, 0, stream>>>((float4*)outa, n4);

    dim3 grid(B_ * H_ * (T_ / TT_));                      // 4096 workgroups
    size_t shmem = (size_t)SM_FLOATS * sizeof(float);     // ~203 KB dynamic LDS
    tamcad_attn_wmma<<<grid, NT_, shmem, stream>>>(q, k, v, ki, vi, outx, outa);
}